// Model_39015482917501
// MI455X (gfx1250) — compile-verified
//
#include <hip/hip_runtime.h>

// ---------------------------------------------------------------------------
// CDNA5 (gfx1250) rasterizing loss kernel.
// w0/w1/z are affine in (px,py) -> evaluated as 16x16x4 fp32 WMMA per
// (16 pixel x 16 face) tile. wave32 throughout.
// ---------------------------------------------------------------------------

typedef __attribute__((ext_vector_type(2))) float v2f;
typedef __attribute__((ext_vector_type(8))) float v8f;

#define IMG      256
#define NPIX     (IMG * IMG)
#define NFACES   768
#define NTILES   (NFACES / 16)   // 48 face tiles
#define TEXD     4

static __device__ __forceinline__ v8f wmma_k4(v2f a, v2f b, v8f c) {
    // D(16x16) = A(16x4) * B(4x16) + C  -- emits v_wmma_f32_16x16x4_f32
    return __builtin_amdgcn_wmma_f32_16x16x4_f32(
        /*neg_a=*/false, a, /*neg_b=*/false, b,
        /*c_mod=*/(short)0, c, /*reuse_a=*/false, /*reuse_b=*/false);
}

// -------------------------------- prep --------------------------------------
// Per face: camera transform + affine coefficients, written directly in the
// B-matrix fragment layout for V_WMMA_F32_16X16X4_F32:
//   lanes 0..15  : (K0,K1) = (A_q, B_q)  for face N = lane
//   lanes 16..31 : (K2,K3) = (C_q, 0)    for face N = lane-16
// Layout: bfrag[(tile*3 + q)*32 + lane]  with q in {w0, w1, z}.
__global__ void prep_kernel(const float* __restrict__ verts,
                            const float* __restrict__ eye,
                            const int* __restrict__ faces,
                            float2* __restrict__ bfrag,
                            int nFaces) {
    int f = blockIdx.x * blockDim.x + threadIdx.x;
    if (f >= nFaces) return;

    float ex = eye[0], ey = eye[1], ez = eye[2];
    float inv = 1.0f / sqrtf(ex * ex + ey * ey + ez * ez);
    float zx = -ex * inv, zy = -ey * inv, zz = -ez * inv;
    // xax = cross(up=(0,1,0), zax) = (zz, 0, -zx), normalized
    float xx = zz, xy = 0.0f, xz = -zx;
    float xinv = 1.0f / sqrtf(xx * xx + xy * xy + xz * xz);
    xx *= xinv; xy *= xinv; xz *= xinv;
    // yax = cross(zax, xax)
    float yx = zy * xz - zz * xy;
    float yy = zz * xx - zx * xz;
    float yz = zx * xy - zy * xx;

    int ia = faces[f * 3 + 0], ib = faces[f * 3 + 1], ic = faces[f * 3 + 2];

    float vax = 0.f, vay = 0.f, vaz = 0.f;
    float vbx = 0.f, vby = 0.f, vbz = 0.f;
    float vcx = 0.f, vcy = 0.f, vcz = 0.f;
#define XFORM(idx, ox, oy, oz)                                   \
    {                                                            \
        float px = verts[(idx) * 3 + 0] - ex;                    \
        float py = verts[(idx) * 3 + 1] - ey;                    \
        float pz = verts[(idx) * 3 + 2] - ez;                    \
        ox = px * xx + py * xy + pz * xz;                        \
        oy = px * yx + py * yy + pz * yz;                        \
        oz = px * zx + py * zy + pz * zz;                        \
    }
    XFORM(ia, vax, vay, vaz)
    XFORM(ib, vbx, vby, vbz)
    XFORM(ic, vcx, vcy, vcz)
#undef XFORM

    float area = (vbx - vax) * (vcy - vay) - (vby - vay) * (vcx - vax);
    float den = (fabsf(area) < 1e-9f) ? 1e-9f : area;
    float iden = 1.0f / den;

    // w0 = ((cx-bx)(py-by) - (cy-by)(px-bx)) / den  ->  A0*px + B0*py + C0
    float A0 = (vby - vcy) * iden;
    float B0 = (vcx - vbx) * iden;
    float C0 = ((vcy - vby) * vbx - (vcx - vbx) * vby) * iden;
    // w1 = ((ax-cx)(py-cy) - (ay-cy)(px-cx)) / den
    float A1 = (vcy - vay) * iden;
    float B1 = (vax - vcx) * iden;
    float C1 = ((vay - vcy) * vcx - (vax - vcx) * vcy) * iden;
    // z = w0*az + w1*bz + (1-w0-w1)*cz
    float dz0 = vaz - vcz, dz1 = vbz - vcz;
    float Az = A0 * dz0 + A1 * dz1;
    float Bz = B0 * dz0 + B1 * dz1;
    float Cz = C0 * dz0 + C1 * dz1 + vcz;

    int tile = f >> 4;
    int lane = f & 15;
    float2* base = bfrag + (size_t)tile * 3 * 32;
    base[0 * 32 + lane]      = make_float2(A0, B0);
    base[0 * 32 + 16 + lane] = make_float2(C0, 0.0f);
    base[1 * 32 + lane]      = make_float2(A1, B1);
    base[1 * 32 + 16 + lane] = make_float2(C1, 0.0f);
    base[2 * 32 + lane]      = make_float2(Az, Bz);
    base[2 * 32 + 16 + lane] = make_float2(Cz, 0.0f);
}

// ------------------------------ rasterize -----------------------------------
static __device__ __forceinline__ unsigned long long
shfl_xor_u64(unsigned long long v, int mask) {
    unsigned lo = (unsigned)v;
    unsigned hi = (unsigned)(v >> 32);
    lo = __shfl_xor(lo, mask, 32);
    hi = __shfl_xor(hi, mask, 32);
    return ((unsigned long long)hi << 32) | lo;
}

__global__ __launch_bounds__(256) void raster_kernel(
        const float2* __restrict__ bfrag,
        const float* __restrict__ tex,       // (NFACES,4,4,4,3)
        const float* __restrict__ img_ref,   // (3,256,256)
        float* __restrict__ out) {
    __shared__ float2 sB[NTILES * 3 * 32];   // 36 KB of B-fragments

    const int tid  = threadIdx.x;
    const int lane = tid & 31;

    for (int i = tid; i < NTILES * 3 * 32; i += 256)
        sB[i] = bfrag[i];
    __syncthreads();

    const int waveG    = blockIdx.x * 8 + (tid >> 5);
    const int pixBase  = waveG * 16;        // 16 pixels per wave

    // A fragment: lanes 0..15 carry (px,py) of pixel M=lane; lanes 16..31
    // carry the constant (K2,K3)=(1,0).
    v2f afrag;
    if (lane < 16) {
        int p = pixBase + lane;
        int row = p >> 8, col = p & 255;
        afrag.x = (col + 0.5f) * (2.0f / IMG) - 1.0f;
        afrag.y = -((row + 0.5f) * (2.0f / IMG) - 1.0f);
    } else {
        afrag.x = 1.0f;
        afrag.y = 0.0f;
    }

    unsigned long long best[8];
#pragma unroll
    for (int m = 0; m < 8; ++m) best[m] = ~0ull;

    const v8f zeroC = {};
    const int nLo = lane & 15;

    for (int t = 0; t < NTILES; ++t) {
        const v2f b0 = *(const v2f*)&sB[(t * 3 + 0) * 32 + lane];
        const v2f b1 = *(const v2f*)&sB[(t * 3 + 1) * 32 + lane];
        const v2f b2 = *(const v2f*)&sB[(t * 3 + 2) * 32 + lane];

        v8f w0 = wmma_k4(afrag, b0, zeroC);
        v8f w1 = wmma_k4(afrag, b1, zeroC);
        v8f zd = wmma_k4(afrag, b2, zeroC);

        unsigned face = (unsigned)(t * 16 + nLo);
#pragma unroll
        for (int m = 0; m < 8; ++m) {
            float W0 = w0[m], W1 = w1[m], Z = zd[m];
            float W2 = 1.0f - W0 - W1;
            bool ok = (W0 >= 0.0f) && (W1 >= 0.0f) && (W2 >= 0.0f) &&
                      (Z > 0.001f);
            unsigned long long key =
                ok ? (((unsigned long long)__float_as_uint(Z) << 32) | face)
                   : ~0ull;
            best[m] = (key < best[m]) ? key : best[m];
        }
    }

    // min-reduce across the 16-lane face dimension (halves stay separate).
#pragma unroll
    for (int mask = 8; mask >= 1; mask >>= 1) {
#pragma unroll
        for (int m = 0; m < 8; ++m) {
            unsigned long long o = shfl_xor_u64(best[m], mask);
            best[m] = (o < best[m]) ? o : best[m];
        }
    }

    // 16 lanes each finish one pixel:
    //   lanes 0..7  -> pixel pixBase + lane        (slots m = lane, lanes<16 own M=m)
    //   lanes 16..23-> pixel pixBase + 8 + (lane-16) (lanes>=16 own M=m+8)
    bool active = (lane < 8) || (lane >= 16 && lane < 24);
    float contrib = 0.0f;
    if (active) {
        int sel = lane & 7;
        unsigned long long k = best[0];
        if (sel == 1) k = best[1];
        if (sel == 2) k = best[2];
        if (sel == 3) k = best[3];
        if (sel == 4) k = best[4];
        if (sel == 5) k = best[5];
        if (sel == 6) k = best[6];
        if (sel == 7) k = best[7];

        int p = pixBase + sel + ((lane >= 16) ? 8 : 0);
        float rr = img_ref[p];
        float rg = img_ref[NPIX + p];
        float rb = img_ref[2 * NPIX + p];

        float cr = 0.0f, cg = 0.0f, cb = 0.0f;
        if ((unsigned)(k >> 32) != 0xFFFFFFFFu) {     // hit
            int f  = (int)(unsigned)k;
            int tt = f >> 4, ll = f & 15;
            float2 c0ab = sB[(tt * 3 + 0) * 32 + ll];
            float2 c0c  = sB[(tt * 3 + 0) * 32 + 16 + ll];
            float2 c1ab = sB[(tt * 3 + 1) * 32 + ll];
            float2 c1c  = sB[(tt * 3 + 1) * 32 + 16 + ll];

            int row = p >> 8, col = p & 255;
            float fx = (col + 0.5f) * (2.0f / IMG) - 1.0f;
            float fy = -((row + 0.5f) * (2.0f / IMG) - 1.0f);

            float W0 = c0ab.x * fx + c0ab.y * fy + c0c.x;
            float W1 = c1ab.x * fx + c1ab.y * fy + c1c.x;
            float W2 = 1.0f - W0 - W1;

            int t0 = (int)rintf(W0 * (TEXD - 1));
            int t1 = (int)rintf(W1 * (TEXD - 1));
            int t2 = (int)rintf(W2 * (TEXD - 1));
            t0 = min(max(t0, 0), TEXD - 1);
            t1 = min(max(t1, 0), TEXD - 1);
            t2 = min(max(t2, 0), TEXD - 1);

            const float* texel =
                tex + ((((size_t)f * TEXD + t0) * TEXD + t1) * TEXD + t2) * 3;
            cr = tanhf(texel[0]);
            cg = tanhf(texel[1]);
            cb = tanhf(texel[2]);
        }
        float d0 = cr - rr, d1 = cg - rg, d2 = cb - rb;
        contrib = d0 * d0 + d1 * d1 + d2 * d2;
    }

    // full-wave sum, one atomic per wave
#pragma unroll
    for (int mask = 16; mask >= 1; mask >>= 1)
        contrib += __shfl_xor(contrib, mask, 32);
    if (lane == 0)
        atomicAdd(out, contrib);
}

// ------------------------------ launcher ------------------------------------
extern "C" void kernel_launch(void* const* d_in, const int* in_sizes, int n_in,
                              void* d_out, int out_size, void* d_ws, size_t ws_size,
                              hipStream_t stream) {
    const float* verts = (const float*)d_in[0];   // (1,408,3)
    const float* tex   = (const float*)d_in[1];   // (1,768,4,4,4,3)
    const float* iref  = (const float*)d_in[2];   // (1,3,256,256)
    const float* eye   = (const float*)d_in[3];   // (3,)
    const int*   faces = (const int*)d_in[4];     // (768,3)
    int nFaces = in_sizes[4] / 3;                 // 768

    float2* bfrag = (float2*)d_ws;                // 48*3*32 float2 = 36 KB
    float*  out   = (float*)d_out;

    hipMemsetAsync(out, 0, sizeof(float), stream);

    prep_kernel<<<(nFaces + 255) / 256, 256, 0, stream>>>(
        verts, eye, faces, bfrag, nFaces);

    // 4096 waves * 16 pixels = 65536 pixels; 8 waves/block -> 512 blocks
    raster_kernel<<<NPIX / 128, 256, 0, stream>>>(bfrag, tex, iref, out);
}